// LIFNode_55130200212118
// MI455X (gfx1250) — compile-verified
//
#include <hip/hip_runtime.h>

// LIF scan over time: X[B, T, N] -> spikes[B, T, N], float32.
//   mem = mem/TAU + x_t ; spike = (mem > 0.5) ; mem = spike ? 0 : mem
// Pure HBM-streaming workload (512 MB total, ~0.27 GFLOP): optimize the
// memory path, not the math. One thread owns 4 consecutive neurons and the
// whole T=16 recurrence for them: 16 independent non-temporal B128 loads
// issued up front, serial scan, 16 non-temporal B128 stores.

typedef float v4f __attribute__((ext_vector_type(4)));

namespace {

constexpr int   T      = 16;
constexpr int   N      = 65536;
constexpr int   N4     = N / 4;        // float4 columns per (b, t) slice
constexpr int   BLOCK  = 256;          // 8 wave32s
constexpr float INV_TAU = 1.0f / 5.0f;
constexpr float V_TH    = 0.5f;

__global__ __launch_bounds__(BLOCK)
void lif_scan_kernel(const v4f* __restrict__ x, v4f* __restrict__ out) {
  // grid: (N4 / BLOCK, B). Lanes of a wave cover 128 consecutive floats of
  // one (b, t) row -> fully coalesced global_load_b128 / global_store_b128.
  const int    n4   = blockIdx.x * BLOCK + threadIdx.x;   // 0 .. N4-1
  const int    b    = blockIdx.y;
  const size_t base = (size_t)b * T * N4 + (size_t)n4;

  // Issue all T loads first: independent streams, hides HBM latency behind
  // the (otherwise serial) recurrence. 16 x v4f = 64 VGPRs of payload.
  v4f xv[T];
#pragma unroll
  for (int t = 0; t < T; ++t) {
    xv[t] = __builtin_nontemporal_load(&x[base + (size_t)t * N4]);
  }

  float m0 = 0.0f, m1 = 0.0f, m2 = 0.0f, m3 = 0.0f;
#pragma unroll
  for (int t = 0; t < T; ++t) {
    m0 = m0 * INV_TAU + xv[t].x;
    m1 = m1 * INV_TAU + xv[t].y;
    m2 = m2 * INV_TAU + xv[t].z;
    m3 = m3 * INV_TAU + xv[t].w;

    const bool s0 = m0 > V_TH;
    const bool s1 = m1 > V_TH;
    const bool s2 = m2 > V_TH;
    const bool s3 = m3 > V_TH;

    v4f sp;
    sp.x = s0 ? 1.0f : 0.0f;
    sp.y = s1 ? 1.0f : 0.0f;
    sp.z = s2 ? 1.0f : 0.0f;
    sp.w = s3 ? 1.0f : 0.0f;

    // hard reset
    m0 = s0 ? 0.0f : m0;
    m1 = s1 ? 0.0f : m1;
    m2 = s2 ? 0.0f : m2;
    m3 = s3 ? 0.0f : m3;

    // Store each timestep immediately: bounds live registers, and spike
    // stores are independent of the remaining recurrence.
    __builtin_nontemporal_store(sp, &out[base + (size_t)t * N4]);
  }
}

} // namespace

extern "C" void kernel_launch(void* const* d_in, const int* in_sizes, int n_in,
                              void* d_out, int out_size, void* d_ws, size_t ws_size,
                              hipStream_t stream) {
  const v4f* X   = (const v4f*)d_in[0];
  v4f*       out = (v4f*)d_out;

  const int total = in_sizes[0];          // B * T * N
  const int B     = total / (T * N);      // 64 for the reference shapes

  dim3 grid(N4 / BLOCK, B);               // (64, 64) = 4096 blocks
  lif_scan_kernel<<<grid, BLOCK, 0, stream>>>(X, out);
}